// MyLLaVA_33930241638410
// MI455X (gfx1250) — compile-verified
//
#include <hip/hip_runtime.h>
#include <stdint.h>

namespace {
constexpr int kB = 8;
constexpr int kT = 2048;
constexpr int kN = 576;
constexpr int kD = 4096;
constexpr int kV = 32000;
constexpr int kImgId = kV - 1;
constexpr int kL = kT + kN - 1;   // 2623
constexpr int kThreads = 256;     // 8 wave32 per block
constexpr int kChunks = kD / (kThreads * 4);  // 4 rounds of float4 per lane
}

// ---------------------------------------------------------------------------
// Kernel 1: find the position of the single IMG_ID token in each row.
// ---------------------------------------------------------------------------
__global__ __launch_bounds__(kThreads) void find_img_pos_kernel(
    const int* __restrict__ ids, int* __restrict__ pos)
{
    __shared__ int s_p;
    const int b = blockIdx.x;
    if (threadIdx.x == 0) s_p = kT;          // sentinel
    __syncthreads();
    for (int t = threadIdx.x; t < kT; t += kThreads) {
        if (ids[b * kT + t] == kImgId) atomicMin(&s_p, t);
    }
    __syncthreads();
    if (threadIdx.x == 0) pos[b] = (s_p == kT) ? 0 : s_p;  // argmax(all-false)==0
}

// ---------------------------------------------------------------------------
// Kernel 2: one block per output row (b, j). Resolve source row, then stream
// 16 KB through LDS with the CDNA5 async global<->LDS DMA path (ASYNCcnt).
// Each lane loads/stores its own LDS chunks, so no barrier is needed between
// the asynccnt wait and the stores. Stores are non-temporal: the 344 MB
// output is write-once, so keep the 192 MB L2 free for embed_table reuse.
// ---------------------------------------------------------------------------
__global__ __launch_bounds__(kThreads) void merge_embed_kernel(
    const int*   __restrict__ ids,   // (B, T)
    const float* __restrict__ img,   // (B, N, D)
    const float* __restrict__ tab,   // (V, D)
    const int*   __restrict__ pos,   // (B,)
    float*       __restrict__ out)   // (B, L, D)
{
    __shared__ float smem[kD];       // 16 KB staging tile
    const int row = blockIdx.x;
    const int b = row / kL;
    const int j = row - b * kL;
    const int p = pos[b];

    const float* src;
    if (j >= p && j < p + kN) {
        src = img + ((size_t)b * kN + (size_t)(j - p)) * kD;
    } else {
        int t = (j < p) ? j : j - (kN - 1);
        t = t < 0 ? 0 : (t > kT - 1 ? kT - 1 : t);
        const int id = ids[b * kT + t];
        src = tab + (size_t)id * kD;
    }
    float* dst = out + (size_t)row * kD;

    const int tid = threadIdx.x;

    // Async copy: global -> LDS (128b per lane per round).
#pragma unroll
    for (int k = 0; k < kChunks; ++k) {
        const int c = k * kThreads + tid;   // 16-byte chunk index
        // Low 32 bits of a flat pointer to LDS == AS(3) byte offset.
        const uint32_t lds_off = (uint32_t)(uintptr_t)(const void*)(smem + c * 4);
        const float* g = src + c * 4;
        asm volatile("global_load_async_to_lds_b128 %0, %1, off"
                     :: "v"(lds_off), "v"(g)
                     : "memory");
    }
    // Wait for this wave's async loads to land in LDS.
    asm volatile("s_wait_asynccnt 0" ::: "memory");

    // Async copy: LDS -> global, non-temporal (streaming, write-once).
#pragma unroll
    for (int k = 0; k < kChunks; ++k) {
        const int c = k * kThreads + tid;
        const uint32_t lds_off = (uint32_t)(uintptr_t)(const void*)(smem + c * 4);
        float* g = dst + c * 4;
        asm volatile("global_store_async_from_lds_b128 %0, %1, off th:TH_STORE_NT"
                     :: "v"(g), "v"(lds_off)
                     : "memory");
    }
    // s_endpgm performs an implicit wait-idle, draining ASYNCcnt before exit.
}

// ---------------------------------------------------------------------------
extern "C" void kernel_launch(void* const* d_in, const int* in_sizes, int n_in,
                              void* d_out, int out_size, void* d_ws, size_t ws_size,
                              hipStream_t stream) {
    const int*   ids = (const int*)d_in[0];    // input_ids  (B*T) int32
    const float* img = (const float*)d_in[1];  // image_embeds (B*N*D) f32
    const float* tab = (const float*)d_in[2];  // embed_table (V*D) f32
    float* out = (float*)d_out;                // (B*L*D) f32
    int*   pos = (int*)d_ws;                   // B ints of scratch

    find_img_pos_kernel<<<kB, kThreads, 0, stream>>>(ids, pos);
    merge_embed_kernel<<<kB * kL, kThreads, 0, stream>>>(ids, img, tab, pos, out);
}